// GraphConv_81423989997747
// MI455X (gfx1250) — compile-verified
//
#include <hip/hip_runtime.h>
#include <hip/hip_bf16.h>

typedef __attribute__((ext_vector_type(2))) float v2f;
typedef __attribute__((ext_vector_type(8))) float v8f;
typedef __attribute__((ext_vector_type(4))) unsigned int u32x4;
typedef __attribute__((ext_vector_type(4))) int i32x4;
typedef __attribute__((ext_vector_type(8))) int i32x8;

// Problem constants (match reference setup_inputs)
constexpr int N_ = 50000;
constexpr int F_ = 256;
constexpr int O_ = 256;
constexpr int R_ = 3;
constexpr int E_ = 1600000;

constexpr int KCH   = 32;           // K-chunk staged in LDS per round
constexpr int LDSW  = 272;          // padded LDS row stride in floats (256 + 16)
constexpr int NTILE = N_ / 16;      // 3125 row tiles

// ---------------------------------------------------------------------------
// Stage a [rows x 256] f32 slab (row-major, row stride 256 floats) from global
// into LDS with +16-float row padding (LDS row stride = 272 floats).
// TDM path: one tensor_load_to_lds issued by wave 0, completion via TENSORcnt.
// Caller must __syncthreads() before (LDS reuse) and after (visibility).
// ---------------------------------------------------------------------------
__device__ __forceinline__ void stage_tile(float* smem, const float* gsrc, int rows) {
#if __has_builtin(__builtin_amdgcn_tensor_load_to_lds)
    if (threadIdx.x < 32) {
        unsigned long long ga = (unsigned long long)(size_t)gsrc;
        unsigned lds_off = (unsigned)(size_t)smem;   // low 32 bits of LDS aperture addr
        u32x4 g0;
        g0[0] = 1u;                                   // count=1, user mode, no gather
        g0[1] = lds_off;                              // lds_addr
        g0[2] = (unsigned)(ga & 0xFFFFFFFFu);         // global_addr[31:0]
        g0[3] = (unsigned)((ga >> 32) & 0x1FFFFFFu)   // global_addr[56:32]
              | (2u << 30);                           // type = 2 ("image")
        i32x8 g1;
        g1[0] = (2 << 16)        // data_size = 4 bytes
              | (1 << 20)        // pad_enable
              | (7 << 22)        // pad_interval: every 256 DWORDs (one row)
              | (15 << 25);      // pad_amount: 16 DWORDs
        g1[1] = (int)(256u << 16);          // tensor_dim0[15:0] = 256
        g1[2] = (int)((unsigned)rows << 16);// tensor_dim0 hi = 0, tensor_dim1 = rows
        g1[3] = (int)(256u << 16);          // tensor_dim1 hi = 0, tile_dim0 = 256
        g1[4] = rows;                       // tile_dim1 = rows, tile_dim2 = 0
        g1[5] = 256;                        // tensor_dim0_stride = 256
        g1[6] = 0;
        g1[7] = 0;
        i32x4 z4 = {0, 0, 0, 0};
#if defined(__clang_major__) && (__clang_major__ >= 23)
        i32x8 z8 = {0, 0, 0, 0, 0, 0, 0, 0};
        __builtin_amdgcn_tensor_load_to_lds(g0, g1, z4, z4, z8, 0);
#else
        __builtin_amdgcn_tensor_load_to_lds(g0, g1, z4, z4, 0);
#endif
        __builtin_amdgcn_s_wait_tensorcnt(0);
    }
#else
    // Fallback: cooperative float4 copy with explicit padding-aware layout.
    for (int q = threadIdx.x; q < rows * 64; q += (int)blockDim.x) {
        int row = q >> 6, c4 = q & 63;
        ((float4*)(smem + row * LDSW))[c4] = ((const float4*)(gsrc + row * 256))[c4];
    }
#endif
}

// ---------------------------------------------------------------------------
// Kernel 1: tiny precompute.
//   W2s[i][j] = W2[2][i][j] + W2[2][256+i][j]        (top+bot fold)
//   M  [i][j] = sum_k B[2][i][k] * W2[2][256+k][j]   (B2 @ W2_bot)
// ---------------------------------------------------------------------------
__global__ void prep_kernel(const float* __restrict__ B,
                            const float* __restrict__ W2,
                            float* __restrict__ W2s,
                            float* __restrict__ Mm) {
    const int i = blockIdx.x;
    const int j = threadIdx.x;
    const float* W2r   = W2 + (size_t)(R_ - 1) * (2 * O_) * O_; // [2O, O]
    const float* W2bot = W2r + (size_t)O_ * O_;
    W2s[i * O_ + j] = W2r[i * O_ + j] + W2bot[i * O_ + j];
    const float* B2 = B + (size_t)(R_ - 1) * F_ * O_;           // [F, O]
    float acc = 0.0f;
    #pragma unroll 4
    for (int k = 0; k < F_; ++k)
        acc = fmaf(B2[i * O_ + k], W2bot[k * O_ + j], acc);
    Mm[i * O_ + j] = acc;
}

// ---------------------------------------------------------------------------
// Kernel 2: zero the aggregation buffer (float4 stores, grid-stride)
// ---------------------------------------------------------------------------
__global__ void zero_kernel(float4* __restrict__ p, long n4) {
    long i = (long)blockIdx.x * blockDim.x + threadIdx.x;
    long stride = (long)gridDim.x * blockDim.x;
    float4 z = {0.f, 0.f, 0.f, 0.f};
    for (; i < n4; i += stride) p[i] = z;
}

// ---------------------------------------------------------------------------
// Kernel 3: h[r] = x @ W[r], f32 WMMA 16x16x4, TDM-staged B in LDS.
// Block = 256 threads = 8 waves = 2 row-tiles x 4 col-groups (32x256 output).
// Grid = (1563, 3); last block's 2nd row tile is guarded.
// ---------------------------------------------------------------------------
__global__ void hgemm_kernel(const float* __restrict__ x,
                             const float* __restrict__ W,
                             float* __restrict__ h) {
    __shared__ float smem[KCH * LDSW];    // 34,816 B

    const int r  = blockIdx.y;
    const int w  = threadIdx.x >> 5;
    const int tn = w & 3;                 // 64-wide column group
    const int tm = blockIdx.x * 2 + (w >> 2);
    const int lane = threadIdx.x & 31;
    const int half = lane >> 4;
    const int lm   = lane & 15;

    int mrow = tm * 16 + lm;
    if (mrow > N_ - 1) mrow = N_ - 1;     // clamp OOB A-rows (stores guarded below)
    const float* xrow = x + (size_t)mrow * F_;
    const float* Wr   = W + (size_t)r * F_ * O_;
    const int nbase   = tn * 64 + lm;

    v8f c0 = {}, c1 = {}, c2 = {}, c3 = {};
    for (int k0 = 0; k0 < F_; k0 += KCH) {
        __syncthreads();                               // protect smem from prev readers
        stage_tile(smem, Wr + (size_t)k0 * O_, KCH);   // TDM: W[k0..k0+31][0:256] -> LDS
        __syncthreads();

        for (int k = 0; k < KCH; k += 4) {
            v2f a;
            a[0] = xrow[k0 + k + 2 * half];
            a[1] = xrow[k0 + k + 2 * half + 1];
            const float* s0 = smem + (k + 2 * half) * LDSW + nbase;
            const float* s1 = s0 + LDSW;
            v2f b0, b1, b2, b3;
            b0[0] = s0[0];  b0[1] = s1[0];
            b1[0] = s0[16]; b1[1] = s1[16];
            b2[0] = s0[32]; b2[1] = s1[32];
            b3[0] = s0[48]; b3[1] = s1[48];
            c0 = __builtin_amdgcn_wmma_f32_16x16x4_f32(false, a, false, b0, (short)0, c0, false, false);
            c1 = __builtin_amdgcn_wmma_f32_16x16x4_f32(false, a, false, b1, (short)0, c1, false, false);
            c2 = __builtin_amdgcn_wmma_f32_16x16x4_f32(false, a, false, b2, (short)0, c2, false, false);
            c3 = __builtin_amdgcn_wmma_f32_16x16x4_f32(false, a, false, b3, (short)0, c3, false, false);
        }
    }

    if (tm < NTILE) {                      // wave-uniform guard (EXEC stays full for WMMA)
        float* outp = h + (size_t)r * N_ * O_ + (size_t)(tm * 16) * O_ + tn * 64 + lm;
        #pragma unroll
        for (int v = 0; v < 8; ++v) {
            size_t ro = (size_t)(v + 8 * half) * O_;
            outp[ro +  0] = c0[v];
            outp[ro + 16] = c1[v];
            outp[ro + 32] = c2[v];
            outp[ro + 48] = c3[v];
        }
    }
}

// ---------------------------------------------------------------------------
// Kernel 4: COO scatter. One wave per edge:
//   agg[row][:] += val * h[r][col][:]   (256 floats, 8 f32 atomics per lane)
// ---------------------------------------------------------------------------
__global__ void scatter_kernel(const int* __restrict__ rows,
                               const int* __restrict__ cols,
                               const float* __restrict__ vals,
                               const float* __restrict__ h,
                               float* __restrict__ agg) {
    const long TOT = (long)R_ * E_;
    long wid = ((long)blockIdx.x * blockDim.x + threadIdx.x) >> 5;
    long nw  = ((long)gridDim.x * blockDim.x) >> 5;
    const int lane = threadIdx.x & 31;

    for (long e = wid; e < TOT; e += nw) {
        const int r   = (int)(e / E_);
        const int row = rows[e];
        const int col = cols[e];
        const float v = vals[e];
        const float4* hp = (const float4*)(h + (size_t)r * N_ * O_ + (size_t)col * O_);
        float* ap = agg + (size_t)row * O_ + lane * 8;
        float4 h0 = hp[lane * 2 + 0];
        float4 h1 = hp[lane * 2 + 1];
        atomicAdd(ap + 0, v * h0.x);
        atomicAdd(ap + 1, v * h0.y);
        atomicAdd(ap + 2, v * h0.z);
        atomicAdd(ap + 3, v * h0.w);
        atomicAdd(ap + 4, v * h1.x);
        atomicAdd(ap + 5, v * h1.y);
        atomicAdd(ap + 6, v * h1.z);
        atomicAdd(ap + 7, v * h1.w);
    }
}

// ---------------------------------------------------------------------------
// Kernel 5: out = relu( agg @ W2s + x @ M ), dual-pass WMMA GEMM with
// TDM-staged B, 32x256 output per block. Grid = 1563.
// ---------------------------------------------------------------------------
__global__ void outgemm_kernel(const float* __restrict__ agg,
                               const float* __restrict__ x,
                               const float* __restrict__ W2s,
                               const float* __restrict__ Mm,
                               float* __restrict__ out) {
    __shared__ float smem[KCH * LDSW];

    const int w  = threadIdx.x >> 5;
    const int tn = w & 3;
    const int tm = blockIdx.x * 2 + (w >> 2);
    const int lane = threadIdx.x & 31;
    const int half = lane >> 4;
    const int lm   = lane & 15;

    int mrow = tm * 16 + lm;
    if (mrow > N_ - 1) mrow = N_ - 1;
    const int nbase = tn * 64 + lm;

    v8f c0 = {}, c1 = {}, c2 = {}, c3 = {};
    for (int pass = 0; pass < 2; ++pass) {
        const float* arow = (pass == 0 ? agg : x) + (size_t)mrow * O_;
        const float* Bm   = (pass == 0 ? W2s : Mm);
        for (int k0 = 0; k0 < O_; k0 += KCH) {
            __syncthreads();
            stage_tile(smem, Bm + (size_t)k0 * O_, KCH);
            __syncthreads();

            for (int k = 0; k < KCH; k += 4) {
                v2f a;
                a[0] = arow[k0 + k + 2 * half];
                a[1] = arow[k0 + k + 2 * half + 1];
                const float* s0 = smem + (k + 2 * half) * LDSW + nbase;
                const float* s1 = s0 + LDSW;
                v2f b0, b1, b2, b3;
                b0[0] = s0[0];  b0[1] = s1[0];
                b1[0] = s0[16]; b1[1] = s1[16];
                b2[0] = s0[32]; b2[1] = s1[32];
                b3[0] = s0[48]; b3[1] = s1[48];
                c0 = __builtin_amdgcn_wmma_f32_16x16x4_f32(false, a, false, b0, (short)0, c0, false, false);
                c1 = __builtin_amdgcn_wmma_f32_16x16x4_f32(false, a, false, b1, (short)0, c1, false, false);
                c2 = __builtin_amdgcn_wmma_f32_16x16x4_f32(false, a, false, b2, (short)0, c2, false, false);
                c3 = __builtin_amdgcn_wmma_f32_16x16x4_f32(false, a, false, b3, (short)0, c3, false, false);
            }
        }
    }

    if (tm < NTILE) {
        float* outp = out + (size_t)(tm * 16) * O_ + tn * 64 + lm;
        #pragma unroll
        for (int v = 0; v < 8; ++v) {
            size_t ro = (size_t)(v + 8 * half) * O_;
            outp[ro +  0] = fmaxf(c0[v], 0.0f);
            outp[ro + 16] = fmaxf(c1[v], 0.0f);
            outp[ro + 32] = fmaxf(c2[v], 0.0f);
            outp[ro + 48] = fmaxf(c3[v], 0.0f);
        }
    }
}

// ---------------------------------------------------------------------------
extern "C" void kernel_launch(void* const* d_in, const int* in_sizes, int n_in,
                              void* d_out, int out_size, void* d_ws, size_t ws_size,
                              hipStream_t stream) {
    const float* x     = (const float*)d_in[0];
    const int*   erows = (const int*)d_in[1];
    const int*   ecols = (const int*)d_in[2];
    const float* evals = (const float*)d_in[3];
    const float* W     = (const float*)d_in[4];
    const float* B     = (const float*)d_in[5];
    const float* W2    = (const float*)d_in[6];
    float* out = (float*)d_out;

    // workspace layout (floats): h[R*N*O] | agg[N*O] | W2s[O*O] | M[O*O]
    float* ws  = (float*)d_ws;
    float* h   = ws;
    float* agg = h + (size_t)R_ * N_ * O_;
    float* W2s = agg + (size_t)N_ * O_;
    float* Mm  = W2s + (size_t)O_ * O_;

    // 1. tiny precompute of folded epilogue matrices
    prep_kernel<<<O_, O_, 0, stream>>>(B, W2, W2s, Mm);

    // 2. zero agg (must happen every call; harness doesn't re-poison)
    zero_kernel<<<4096, 256, 0, stream>>>((float4*)agg, (long)N_ * O_ / 4);

    // 3. h[r] = x @ W[r]  (TDM-staged B, 32x256 per block)
    {
        dim3 grid((NTILE + 1) / 2, R_);
        hgemm_kernel<<<grid, 256, 0, stream>>>(x, W, h);
    }

    // 4. edge scatter (memory/atomic bound — the real bottleneck)
    scatter_kernel<<<16384, 256, 0, stream>>>(erows, ecols, evals, h, agg);

    // 5. fused epilogue GEMM + relu
    outgemm_kernel<<<(NTILE + 1) / 2, 256, 0, stream>>>(agg, x, W2s, Mm, out);
}